// ListNetRankingLoss_73005854097661
// MI455X (gfx1250) — compile-verified
//
#include <hip/hip_runtime.h>
#include <math.h>

// ListNet ranking loss, MI455X (gfx1250, wave32).
// Memory-bound: 64 MiB input -> ~2.9 us floor at 23.3 TB/s.
// One wave per row; row staged in LDS via CDNA5 async global->LDS copies.

constexpr int KMAX = 512;            // Kmax of this problem
constexpr int ROWS_PER_BLOCK = 8;    // 8 waves per block, one row each
constexpr int THREADS = ROWS_PER_BLOCK * 32;

// Fast path: K = NCHUNK * 128, fully unrolled, async global->LDS staging.
template <int NCHUNK>
__global__ __launch_bounds__(THREADS) void listnet_rows_kernel(
    const float* __restrict__ pred,
    const float* __restrict__ tgt,
    const int*   __restrict__ lengths,
    float* __restrict__ row_loss,
    float* __restrict__ row_valid,
    int B)
{
    constexpr int K = NCHUNK * 128;
    __shared__ __align__(16) float smem[ROWS_PER_BLOCK][2 * KMAX];

    const int wave = threadIdx.x >> 5;
    const int lane = threadIdx.x & 31;
    const int r    = blockIdx.x * ROWS_PER_BLOCK + wave;
    if (r >= B) return;

    const int len = lengths[r];
    const float* prow = pred + (size_t)r * K;
    const float* trow = tgt  + (size_t)r * K;

    // ---- CDNA5 async copy: global -> LDS, tracked by ASYNCcnt ----
    #pragma unroll
    for (int c = 0; c < NCHUNK; ++c) {
        const int e = (c * 32 + lane) * 4;  // element index of this lane's float4
        unsigned ldsp = (unsigned)(uintptr_t)&smem[wave][e];
        unsigned ldst = (unsigned)(uintptr_t)&smem[wave][KMAX + e];
        const float* gp = prow + e;
        const float* gt = trow + e;
        asm volatile("global_load_async_to_lds_b128 %0, %1, off"
                     :: "v"(ldsp), "v"(gp) : "memory");
        asm volatile("global_load_async_to_lds_b128 %0, %1, off"
                     :: "v"(ldst), "v"(gt) : "memory");
    }
    asm volatile("s_wait_asynccnt 0x0" ::: "memory");

    // ---- read row back from LDS into registers (ds_load_b128) ----
    float Pf[4 * NCHUNK], Tf[4 * NCHUNK];
    float mp = -INFINITY, mt = -INFINITY;
    #pragma unroll
    for (int c = 0; c < NCHUNK; ++c) {
        const int e = (c * 32 + lane) * 4;
        const float4 p4 = *reinterpret_cast<const float4*>(&smem[wave][e]);
        const float4 t4 = *reinterpret_cast<const float4*>(&smem[wave][KMAX + e]);
        Pf[c * 4 + 0] = p4.x; Pf[c * 4 + 1] = p4.y;
        Pf[c * 4 + 2] = p4.z; Pf[c * 4 + 3] = p4.w;
        Tf[c * 4 + 0] = t4.x; Tf[c * 4 + 1] = t4.y;
        Tf[c * 4 + 2] = t4.z; Tf[c * 4 + 3] = t4.w;
        #pragma unroll
        for (int j = 0; j < 4; ++j) {
            if (e + j < len) {
                mp = fmaxf(mp,  Pf[c * 4 + j]);
                mt = fmaxf(mt, -Tf[c * 4 + j]);
            }
        }
    }

    // wave32 max-reduce
    #pragma unroll
    for (int off = 16; off > 0; off >>= 1) {
        mp = fmaxf(mp, __shfl_xor(mp, off, 32));
        mt = fmaxf(mt, __shfl_xor(mt, off, 32));
    }

    // sums: sp = sum exp(p-mp); st = sum exp(-t-mt); dot = sum exp(-t-mt)*p
    float sp = 0.f, st = 0.f, dot = 0.f;
    #pragma unroll
    for (int i = 0; i < 4 * NCHUNK; ++i) {
        const int c = i >> 2, j = i & 3;
        const int idx = (c * 32 + lane) * 4 + j;
        if (idx < len) {
            sp += __expf(Pf[i] - mp);
            const float ex = __expf(-Tf[i] - mt);
            st  += ex;
            dot += ex * Pf[i];
        }
    }
    #pragma unroll
    for (int off = 16; off > 0; off >>= 1) {
        sp  += __shfl_xor(sp,  off, 32);
        st  += __shfl_xor(st,  off, 32);
        dot += __shfl_xor(dot, off, 32);
    }

    if (lane == 0) {
        const bool ok = (len >= 2);
        float loss = 0.f;
        if (ok) loss = (mp + __logf(sp)) - dot / st;  // logZ - E_tp[pred]
        row_loss[r]  = loss;
        row_valid[r] = ok ? 1.f : 0.f;
    }
}

// Generic fallback (any K): one wave per row, two passes (row stays L2-hot).
__global__ __launch_bounds__(THREADS) void listnet_rows_generic(
    const float* __restrict__ pred,
    const float* __restrict__ tgt,
    const int*   __restrict__ lengths,
    float* __restrict__ row_loss,
    float* __restrict__ row_valid,
    int B, int K)
{
    const int wave = threadIdx.x >> 5;
    const int lane = threadIdx.x & 31;
    const int r    = blockIdx.x * ROWS_PER_BLOCK + wave;
    if (r >= B) return;

    const int len = lengths[r];
    const float* prow = pred + (size_t)r * K;
    const float* trow = tgt  + (size_t)r * K;

    float mp = -INFINITY, mt = -INFINITY;
    for (int i = lane; i < len; i += 32) {
        mp = fmaxf(mp,  prow[i]);
        mt = fmaxf(mt, -trow[i]);
    }
    #pragma unroll
    for (int off = 16; off > 0; off >>= 1) {
        mp = fmaxf(mp, __shfl_xor(mp, off, 32));
        mt = fmaxf(mt, __shfl_xor(mt, off, 32));
    }
    float sp = 0.f, st = 0.f, dot = 0.f;
    for (int i = lane; i < len; i += 32) {
        const float p = prow[i];
        sp += __expf(p - mp);
        const float ex = __expf(-trow[i] - mt);
        st  += ex;
        dot += ex * p;
    }
    #pragma unroll
    for (int off = 16; off > 0; off >>= 1) {
        sp  += __shfl_xor(sp,  off, 32);
        st  += __shfl_xor(st,  off, 32);
        dot += __shfl_xor(dot, off, 32);
    }
    if (lane == 0) {
        const bool ok = (len >= 2);
        float loss = 0.f;
        if (ok) loss = (mp + __logf(sp)) - dot / st;
        row_loss[r]  = loss;
        row_valid[r] = ok ? 1.f : 0.f;
    }
}

// Deterministic final reduction: one block sums all per-row results.
__global__ __launch_bounds__(1024) void listnet_final_kernel(
    const float* __restrict__ row_loss,
    const float* __restrict__ row_valid,
    float* __restrict__ out, int B)
{
    __shared__ float sl[32];
    __shared__ float sv[32];

    float a = 0.f, b = 0.f;
    for (int i = threadIdx.x; i < B; i += 1024) {
        a += row_loss[i];
        b += row_valid[i];
    }
    #pragma unroll
    for (int off = 16; off > 0; off >>= 1) {
        a += __shfl_xor(a, off, 32);
        b += __shfl_xor(b, off, 32);
    }
    const int wave = threadIdx.x >> 5;
    const int lane = threadIdx.x & 31;
    if (lane == 0) { sl[wave] = a; sv[wave] = b; }
    __syncthreads();
    if (threadIdx.x < 32) {
        a = sl[threadIdx.x];
        b = sv[threadIdx.x];
        #pragma unroll
        for (int off = 16; off > 0; off >>= 1) {
            a += __shfl_xor(a, off, 32);
            b += __shfl_xor(b, off, 32);
        }
        if (threadIdx.x == 0)
            out[0] = (b > 0.f) ? (a / fmaxf(b, 1.0f)) : 0.0f;
    }
}

extern "C" void kernel_launch(void* const* d_in, const int* in_sizes, int n_in,
                              void* d_out, int out_size, void* d_ws, size_t ws_size,
                              hipStream_t stream) {
    const float* pred    = (const float*)d_in[0];
    const float* tgt     = (const float*)d_in[1];
    const int*   lengths = (const int*)d_in[2];

    const int B = in_sizes[2];            // 16384
    const int K = in_sizes[0] / B;        // 512

    float* row_loss  = (float*)d_ws;      // B floats
    float* row_valid = row_loss + B;      // B floats  (128 KiB total)

    const int blocks = (B + ROWS_PER_BLOCK - 1) / ROWS_PER_BLOCK;
    switch (K) {
    case 512:
        listnet_rows_kernel<4><<<blocks, THREADS, 0, stream>>>(
            pred, tgt, lengths, row_loss, row_valid, B);
        break;
    case 384:
        listnet_rows_kernel<3><<<blocks, THREADS, 0, stream>>>(
            pred, tgt, lengths, row_loss, row_valid, B);
        break;
    case 256:
        listnet_rows_kernel<2><<<blocks, THREADS, 0, stream>>>(
            pred, tgt, lengths, row_loss, row_valid, B);
        break;
    case 128:
        listnet_rows_kernel<1><<<blocks, THREADS, 0, stream>>>(
            pred, tgt, lengths, row_loss, row_valid, B);
        break;
    default:
        listnet_rows_generic<<<blocks, THREADS, 0, stream>>>(
            pred, tgt, lengths, row_loss, row_valid, B, K);
        break;
    }
    listnet_final_kernel<<<1, 1024, 0, stream>>>(
        row_loss, row_valid, (float*)d_out, B);
}